// OpticFlow_61684320305781
// MI455X (gfx1250) — compile-verified
//
#include <hip/hip_runtime.h>
#include <math.h>

// ---------------------------------------------------------------------------
// MI455X (gfx1250) CLG optical-flow multigrid.
//   * All intermediates live in d_ws (~173 MB) -> resident in 192 MB L2.
//   * sigma=2.0 13-tap separable blur done with V_WMMA_F32_16X16X4_F32
//     (full-precision f32 matrix op): OUT(16x16) = IN(16x28) x S(28x16).
//     Constant banded-matrix fragments hoisted out of the tile loop;
//     interior tiles use branch-free 8B vector loads.
//   * Everything else: fused L2-resident stencil sweeps, wave32-friendly
//     256-thread blocks.
// ---------------------------------------------------------------------------

typedef __attribute__((ext_vector_type(2))) float v2f;
typedef __attribute__((ext_vector_type(8))) float v8f;

// 13 Gaussian taps (sigma=2) embedded in a zero-padded table at [24..36] so
// any shifted index stays in [0,63] -> unconditional indexed reads.
__device__ __constant__ float c_k13p[64] = {
    0.f, 0.f, 0.f, 0.f, 0.f, 0.f, 0.f, 0.f, 0.f, 0.f, 0.f, 0.f,
    0.f, 0.f, 0.f, 0.f, 0.f, 0.f, 0.f, 0.f, 0.f, 0.f, 0.f, 0.f,
    0.00221820f, 0.00877313f, 0.02702317f, 0.06482518f, 0.12110939f,
    0.17621325f, 0.19967566f, 0.17621325f, 0.12110939f, 0.06482518f,
    0.02702317f, 0.00877313f, 0.00221820f,
    0.f, 0.f, 0.f, 0.f, 0.f, 0.f, 0.f, 0.f, 0.f, 0.f, 0.f, 0.f, 0.f, 0.f,
    0.f, 0.f, 0.f, 0.f, 0.f, 0.f, 0.f, 0.f, 0.f, 0.f, 0.f, 0.f, 0.f};
__device__ __constant__ float c_k5[5] = {
    0.00026387f, 0.10645111f, 0.78657004f, 0.10645111f, 0.00026387f};

#define TPW 4  // tiles per wave in the WMMA blur kernels

// ---------------------------------------------------------------------------
// 13-tap horizontal blur via f32 WMMA. One wave = TPW 16x16 tiles.
// OUT = A(16x28) * S(28x16), S[j][n] = k13[n+6-j] (banded, constant).
// A layout (32-bit A 16x4): lane L: m=L&15, half=L>>4; v0=K(4c+2h), v1=K(+1).
// B layout mirrored; C/D: VGPR i -> row (i + 8*half), col (L&15).
// ---------------------------------------------------------------------------
__global__ __launch_bounds__(256) void blur13_h_wmma(
    const float* __restrict__ in, float* __restrict__ out, int H, int W) {
  const int lane = threadIdx.x & 31;
  const int wave = threadIdx.x >> 5;
  const int ty = blockIdx.y;
  const int p = blockIdx.z;
  const size_t plane = (size_t)p * H * W;
  const float* __restrict__ src = in + plane;
  float* __restrict__ dst = out + plane;
  const int m = lane & 15, half = lane >> 4;
  const int row = ty * 16 + m;  // A-matrix row M (H multiple of 16)
  const float* __restrict__ srow = src + (size_t)row * W;

  // Tile-independent B fragments (banded Gaussian matrix), branch-free.
  v2f bf[7];
#pragma unroll
  for (int c = 0; c < 7; ++c) {
    const int j0 = 4 * c + 2 * half;
    bf[c].x = c_k13p[30 + m - j0];
    bf[c].y = c_k13p[29 + m - j0];
  }

  const int tx0 = (blockIdx.x * 8 + wave) * TPW;
#pragma unroll
  for (int t = 0; t < TPW; ++t) {
    const int x0 = (tx0 + t) * 16;
    v8f acc = {0.f, 0.f, 0.f, 0.f, 0.f, 0.f, 0.f, 0.f};
    if (x0 >= 6 && x0 + 21 < W) {  // interior (wave-uniform): raw 8B loads
#pragma unroll
      for (int c = 0; c < 7; ++c) {
        const int xa = x0 - 6 + 4 * c + 2 * half;  // even -> 8B aligned
        const v2f a = *(const v2f*)(srow + xa);
        acc = __builtin_amdgcn_wmma_f32_16x16x4_f32(false, a, false, bf[c],
                                                    (short)0, acc, false, false);
      }
    } else {  // edge tiles: clamp index + 0/1 mask (cndmask, no exec branch)
#pragma unroll
      for (int c = 0; c < 7; ++c) {
        const int xa = x0 - 6 + 4 * c + 2 * half;
        const int ia = min(max(xa, 0), W - 1);
        const int ib = min(max(xa + 1, 0), W - 1);
        v2f a;
        a.x = srow[ia] * ((xa >= 0 && xa < W) ? 1.f : 0.f);
        a.y = srow[ib] * ((xa + 1 >= 0 && xa + 1 < W) ? 1.f : 0.f);
        acc = __builtin_amdgcn_wmma_f32_16x16x4_f32(false, a, false, bf[c],
                                                    (short)0, acc, false, false);
      }
    }
#pragma unroll
    for (int i = 0; i < 8; ++i)
      dst[(size_t)(ty * 16 + i + 8 * half) * W + x0 + m] = acc[i];
  }
}

// Vertical pass: OUT = Sv(16x28) * B(28x16), Sv[m][j] = k13[j-m].
__global__ __launch_bounds__(256) void blur13_v_wmma(
    const float* __restrict__ in, float* __restrict__ out, int H, int W) {
  const int lane = threadIdx.x & 31;
  const int wave = threadIdx.x >> 5;
  const int ty = blockIdx.y;
  const int p = blockIdx.z;
  const size_t plane = (size_t)p * H * W;
  const float* __restrict__ src = in + plane;
  float* __restrict__ dst = out + plane;
  const int m = lane & 15, half = lane >> 4;
  const int y0 = ty * 16;

  // Tile-independent A fragments (transposed banded matrix), branch-free.
  v2f af[7];
#pragma unroll
  for (int c = 0; c < 7; ++c) {
    const int j0 = 4 * c + 2 * half;
    af[c].x = c_k13p[24 + j0 - m];
    af[c].y = c_k13p[25 + j0 - m];
  }

  const int tx0 = (blockIdx.x * 8 + wave) * TPW;
  const bool interior = (y0 >= 6) && (y0 + 21 < H);  // block-uniform
#pragma unroll
  for (int t = 0; t < TPW; ++t) {
    const int x0 = (tx0 + t) * 16;
    v8f acc = {0.f, 0.f, 0.f, 0.f, 0.f, 0.f, 0.f, 0.f};
    if (interior) {
#pragma unroll
      for (int c = 0; c < 7; ++c) {
        const int ry = y0 - 6 + 4 * c + 2 * half;
        v2f b;
        b.x = src[(size_t)ry * W + x0 + m];
        b.y = src[(size_t)(ry + 1) * W + x0 + m];
        acc = __builtin_amdgcn_wmma_f32_16x16x4_f32(false, af[c], false, b,
                                                    (short)0, acc, false, false);
      }
    } else {
#pragma unroll
      for (int c = 0; c < 7; ++c) {
        const int ry = y0 - 6 + 4 * c + 2 * half;
        const int ra = min(max(ry, 0), H - 1);
        const int rb = min(max(ry + 1, 0), H - 1);
        v2f b;
        b.x = src[(size_t)ra * W + x0 + m] * ((ry >= 0 && ry < H) ? 1.f : 0.f);
        b.y = src[(size_t)rb * W + x0 + m] * ((ry + 1 >= 0 && ry + 1 < H) ? 1.f : 0.f);
        acc = __builtin_amdgcn_wmma_f32_16x16x4_f32(false, af[c], false, b,
                                                    (short)0, acc, false, false);
      }
    }
#pragma unroll
    for (int i = 0; i < 8; ++i)
      dst[(size_t)(y0 + i + 8 * half) * W + x0 + m] = acc[i];
  }
}

// ---------------------------------------------------------------------------
// 5-tap separable blur (sigma=0.5), LDS-tiled, zero padding.
// ---------------------------------------------------------------------------
__global__ __launch_bounds__(256) void blur5_kernel(
    const float* __restrict__ in, float* __restrict__ out, int H, int W) {
  __shared__ float tin[20][20];
  __shared__ float th[20][16];
  const int p = blockIdx.z;
  const size_t plane = (size_t)p * H * W;
  const float* __restrict__ src = in + plane;
  float* __restrict__ dst = out + plane;
  const int x0 = blockIdx.x * 16, y0 = blockIdx.y * 16;

  for (int idx = threadIdx.x; idx < 400; idx += 256) {
    const int ly = idx / 20, lx = idx - ly * 20;
    const int gx = x0 - 2 + lx, gy = y0 - 2 + ly;
    tin[ly][lx] = (gx >= 0 && gx < W && gy >= 0 && gy < H)
                      ? src[(size_t)gy * W + gx] : 0.f;
  }
  __syncthreads();
  for (int idx = threadIdx.x; idx < 320; idx += 256) {
    const int ly = idx / 16, lx = idx - ly * 16;
    float s = 0.f;
#pragma unroll
    for (int t = 0; t < 5; ++t) s += c_k5[t] * tin[ly][lx + t];
    th[ly][lx] = s;
  }
  __syncthreads();
  const int lx = threadIdx.x & 15, ly = threadIdx.x >> 4;
  float s = 0.f;
#pragma unroll
  for (int t = 0; t < 5; ++t) s += c_k5[t] * th[ly + t][lx];
  dst[(size_t)(y0 + ly) * W + x0 + lx] = s;
}

// ---------------------------------------------------------------------------
// Fused: central diffs (replicate edges) -> tensor products summed over the
// 3 channels -> P[field*2+b] planes (J11,J12,J13,J22,J23) + diffusivity g.
// ---------------------------------------------------------------------------
__global__ __launch_bounds__(256) void struct_products(
    const float* __restrict__ f1b, const float* __restrict__ f2b,
    float* __restrict__ P, float* __restrict__ g, int H, int W) {
  const int x = blockIdx.x * blockDim.x + threadIdx.x;
  const int y = blockIdx.y;
  const int b = blockIdx.z;
  if (x >= W) return;
  const int xe = min(x + 1, W - 1), xw = max(x - 1, 0);
  const int yn = max(y - 1, 0), ys = min(y + 1, H - 1);
  const size_t HWl = (size_t)H * W;
  float J11 = 0.f, J12 = 0.f, J13 = 0.f, J22 = 0.f, J23 = 0.f, gs = 0.f;
#pragma unroll
  for (int c = 0; c < 3; ++c) {
    const float* __restrict__ a = f1b + (size_t)(b * 3 + c) * HWl;
    const float* __restrict__ bb = f2b + (size_t)(b * 3 + c) * HWl;
    const size_t ro = (size_t)y * W, rn = (size_t)yn * W, rs = (size_t)ys * W;
    const float fx1 = 0.5f * (a[ro + xe] - a[ro + xw]);
    const float fy1 = 0.5f * (a[rs + x] - a[rn + x]);
    const float fx2 = 0.5f * (bb[ro + xe] - bb[ro + xw]);
    const float fy2 = 0.5f * (bb[rs + x] - bb[rn + x]);
    const float nfx = 0.5f * (fx1 + fx2);
    const float nfy = 0.5f * (fy1 + fy2);
    const float nft = bb[ro + x] - a[ro + x];
    J11 += nfx * nfx; J12 += nfx * nfy; J13 += nfx * nft;
    J22 += nfy * nfy; J23 += nfy * nft;
    gs += nfx * nfx + nfy * nfy;
  }
  const size_t o = (size_t)y * W + x;
  P[(0 * 2 + b) * HWl + o] = J11;
  P[(1 * 2 + b) * HWl + o] = J12;
  P[(2 * 2 + b) * HWl + o] = J13;
  P[(3 * 2 + b) * HWl + o] = J22;
  P[(4 * 2 + b) * HWl + o] = J23;
  g[(size_t)b * HWl + o] = rsqrtf(1.0f + gs * (1.0f / 25.0f));  // lambda=5
}

// ---------------------------------------------------------------------------
// Combined diffusion-term stencil: mode 0 = elliptic Jacobi, 1 = parabolic
// Jacobi (tau=0.2), 2 = residual. Replicate (Neumann) boundaries.
// ---------------------------------------------------------------------------
__global__ __launch_bounds__(256) void cdt_kernel(
    const float* __restrict__ U, const float* __restrict__ V,
    const float* __restrict__ J11p, const float* __restrict__ J12p,
    const float* __restrict__ J13p, const float* __restrict__ J22p,
    const float* __restrict__ J23p, const float* __restrict__ G,
    float* __restrict__ OU, float* __restrict__ OV, int h, int w,
    float inv_hx2, int mode) {
  const int x = blockIdx.x * blockDim.x + threadIdx.x;
  const int y = blockIdx.y;
  const int b = blockIdx.z;
  if (x >= w) return;
  const size_t hw = (size_t)h * w, pl = (size_t)b * hw;
  const float* __restrict__ u = U + pl;
  const float* __restrict__ v = V + pl;
  const float* __restrict__ g = G + pl;
  const int xe = min(x + 1, w - 1), xw = max(x - 1, 0);
  const int yn = max(y - 1, 0), ys = min(y + 1, h - 1);
  const size_t ro = (size_t)y * w, rn = (size_t)yn * w, rs = (size_t)ys * w;
  const float gc = g[ro + x];
  const float nhx = 0.5f * (g[ro + xe] + gc), phx = 0.5f * (g[ro + xw] + gc);
  const float nhy = 0.5f * (g[rs + x] + gc), phy = 0.5f * (g[rn + x] + gc);
  const float snu = inv_hx2 * (nhx * u[ro + xe] + phx * u[ro + xw] +
                               nhy * u[rs + x] + phy * u[rn + x]);
  const float snv = inv_hx2 * (nhx * v[ro + xe] + phx * v[ro + xw] +
                               nhy * v[rs + x] + phy * v[rn + x]);
  const float cw = inv_hx2 * (nhx + phx + nhy + phy);
  const float uc = u[ro + x], vc = v[ro + x];
  const size_t o = pl + ro + x;
  const float j11 = J11p[o], j12 = J12p[o], j13 = J13p[o];
  const float j22 = J22p[o], j23 = J23p[o];
  const float factor = 0.002f;  // 1/alpha
  if (mode == 0) {
    OU[o] = (factor * (j12 * vc + j13) - snu) / (factor * j11 - cw);
    OV[o] = (factor * (j12 * uc + j23) - snv) / (factor * j22 - cw);
  } else {
    const float ru = factor * (j12 * vc + j13) - (snu - cw * uc + factor * j11 * uc);
    const float rv = factor * (j12 * uc + j23) - (snv - cw * vc + factor * j22 * vc);
    if (mode == 1) { OU[o] = uc + 0.2f * ru; OV[o] = vc + 0.2f * rv; }
    else           { OU[o] = ru;             OV[o] = rv; }
  }
}

// ---------------------------------------------------------------------------
// Restriction: maxpool2 of J11,J12,J22 + residuals (into J13/J23 slots),
// avgpool2 of g.  Coarse J layout matches fine: field*2*hwc + b*hwc.
// ---------------------------------------------------------------------------
__global__ __launch_bounds__(256) void restrict_kernel(
    const float* __restrict__ J11f, const float* __restrict__ J12f,
    const float* __restrict__ J22f, const float* __restrict__ ruf,
    const float* __restrict__ rvf, const float* __restrict__ gf,
    float* __restrict__ Jc, float* __restrict__ gc, int hc, int wc) {
  const int x = blockIdx.x * blockDim.x + threadIdx.x;
  const int y = blockIdx.y;
  const int b = blockIdx.z;
  if (x >= wc) return;
  const int wf = wc * 2;
  const size_t hwf = (size_t)hc * 2 * wf, hwc = (size_t)hc * wc;
  const size_t i00 = (size_t)b * hwf + (size_t)(2 * y) * wf + 2 * x;
  const size_t o = (size_t)b * hwc + (size_t)y * wc + x;
#define MAX4(f) fmaxf(fmaxf(f[i00], f[i00 + 1]), fmaxf(f[i00 + wf], f[i00 + wf + 1]))
  Jc[0 * 2 * hwc + o] = MAX4(J11f);
  Jc[1 * 2 * hwc + o] = MAX4(J12f);
  Jc[2 * 2 * hwc + o] = MAX4(ruf);
  Jc[3 * 2 * hwc + o] = MAX4(J22f);
  Jc[4 * 2 * hwc + o] = MAX4(rvf);
#undef MAX4
  gc[o] = 0.25f * (gf[i00] + gf[i00 + 1] + gf[i00 + wf] + gf[i00 + wf + 1]);
}

// Bilinear x2 prolongation (half-pixel centers, edge clamp) added into u.
__global__ __launch_bounds__(256) void upsample_add(
    const float* __restrict__ e, float* __restrict__ u, int hf, int wf) {
  const int x = blockIdx.x * blockDim.x + threadIdx.x;
  const int y = blockIdx.y;
  const int b = blockIdx.z;
  if (x >= wf) return;
  const int hc = hf >> 1, wc = wf >> 1;
  const size_t co = (size_t)b * hc * wc;
  const float sy = (y + 0.5f) * 0.5f - 0.5f;
  const float sx = (x + 0.5f) * 0.5f - 0.5f;
  const int y0 = (int)floorf(sy), x0 = (int)floorf(sx);
  const float wy = sy - (float)y0, wx = sx - (float)x0;
  const int y0c = max(0, min(hc - 1, y0)), y1c = max(0, min(hc - 1, y0 + 1));
  const int x0c = max(0, min(wc - 1, x0)), x1c = max(0, min(wc - 1, x0 + 1));
  const float v00 = e[co + (size_t)y0c * wc + x0c];
  const float v01 = e[co + (size_t)y0c * wc + x1c];
  const float v10 = e[co + (size_t)y1c * wc + x0c];
  const float v11 = e[co + (size_t)y1c * wc + x1c];
  const float val = (1.f - wy) * ((1.f - wx) * v00 + wx * v01) +
                    wy * ((1.f - wx) * v10 + wx * v11);
  u[(size_t)b * hf * wf + (size_t)y * wf + x] += val;
}

__global__ void fill_zero(float* __restrict__ p, size_t n) {
  const size_t i = (size_t)blockIdx.x * 256 + threadIdx.x;
  if (i < n) p[i] = 0.f;
}

static inline void swapf(float*& a, float*& b) { float* t = a; a = b; b = t; }

extern "C" void kernel_launch(void* const* d_in, const int* in_sizes, int n_in,
                              void* d_out, int out_size, void* d_ws, size_t ws_size,
                              hipStream_t stream) {
  (void)in_sizes; (void)n_in; (void)out_size;
  const float* f1 = (const float*)d_in[0];
  const float* f2 = (const float*)d_in[1];
  float* out = (float*)d_out;

  constexpr int Bt = 2, H = 1024, W = 1024;
  const size_t HW = (size_t)H * W;
  const size_t HW2 = HW / 4, HW3 = HW / 16;

  float* ws = (float*)d_ws;
  size_t off = 0;
  auto alloc = [&](size_t n) { float* p = ws + off; off += n; return p; };

  float* f1b = alloc(Bt * 3 * HW);   // 6M  (also reused as blur-temp T)
  float* f2b = alloc(Bt * 3 * HW);   // 6M
  float* P   = alloc(10 * HW);       // 10M (also reused as final J1)
  float* g1  = alloc(Bt * HW);
  float* u   = alloc(Bt * HW); float* v  = alloc(Bt * HW);
  float* u2  = alloc(Bt * HW); float* v2 = alloc(Bt * HW);
  float* ru  = alloc(Bt * HW); float* rv = alloc(Bt * HW);
  float* J2  = alloc(10 * HW2); float* g2 = alloc(Bt * HW2);
  float* e1  = alloc(Bt * HW2); float* e2 = alloc(Bt * HW2);
  float* e1b = alloc(Bt * HW2); float* e2b = alloc(Bt * HW2);
  float* r2u = alloc(Bt * HW2); float* r2v = alloc(Bt * HW2);
  float* J3  = alloc(10 * HW3); float* g3 = alloc(Bt * HW3);
  float* e31 = alloc(Bt * HW3); float* e32 = alloc(Bt * HW3);
  float* e31b = alloc(Bt * HW3); float* e32b = alloc(Bt * HW3);
  if (ws_size < off * sizeof(float)) return;  // needs ~173 MB scratch

  float* T  = f1b;  // 10 planes fit in the 12-plane f1b+f2b region
  float* J1 = P;    // vertical pass reads T, writes over dead P

  auto zero = [&](float* p, size_t n) {
    fill_zero<<<dim3((unsigned)((n + 255) / 256)), 256, 0, stream>>>(p, n);
  };
  auto launch_cdt = [&](const float* U, const float* V, float* Jb,
                        const float* G, float* OU, float* OV, int h, int w,
                        float inv_hx2, int mode) {
    const size_t hw = (size_t)h * w;
    dim3 g((w + 255) / 256, h, Bt);
    cdt_kernel<<<g, 256, 0, stream>>>(U, V, Jb, Jb + 2 * hw, Jb + 4 * hw,
                                      Jb + 6 * hw, Jb + 8 * hw, G, OU, OV, h,
                                      w, inv_hx2, mode);
  };
  auto jacobi2 = [&](float*& U, float*& V, float*& Ub, float*& Vb, float* Jb,
                     const float* G, int h, int w, float inv_hx2, int mode) {
    for (int i = 0; i < 2; ++i) {  // SMOOTH_ITER=2 (even -> pointers stable)
      launch_cdt(U, V, Jb, G, Ub, Vb, h, w, inv_hx2, mode);
      swapf(U, Ub); swapf(V, Vb);
    }
  };

  // ---- structure tensor ----
  dim3 gB5(W / 16, H / 16, Bt * 3);
  blur5_kernel<<<gB5, 256, 0, stream>>>(f1, f1b, H, W);
  blur5_kernel<<<gB5, 256, 0, stream>>>(f2, f2b, H, W);
  dim3 gP((W + 255) / 256, H, Bt);
  struct_products<<<gP, 256, 0, stream>>>(f1b, f2b, P, g1, H, W);
  // 8 waves/block, TPW=4 tiles per wave -> grid.x = (W/16)/(8*TPW) = 2
  dim3 gWm(W / 16 / (8 * TPW), H / 16, 10);
  blur13_h_wmma<<<gWm, 256, 0, stream>>>(P, T, H, W);
  blur13_v_wmma<<<gWm, 256, 0, stream>>>(T, J1, H, W);

  // ---- multigrid ----
  zero(u, Bt * HW); zero(v, Bt * HW);
  for (int cyc = 0; cyc < 3; ++cyc) {
    // depth 1 (1024^2, hx=1)
    jacobi2(u, v, u2, v2, J1, g1, H, W, 1.0f, 0);
    launch_cdt(u, v, J1, g1, ru, rv, H, W, 1.0f, 2);
    { dim3 gr((512 + 255) / 256, 512, Bt);
      restrict_kernel<<<gr, 256, 0, stream>>>(J1, J1 + 2 * HW, J1 + 6 * HW,
                                              ru, rv, g1, J2, g2, 512, 512); }
    zero(e1, Bt * HW2); zero(e2, Bt * HW2);
    // depth 2 (512^2, hx=2 -> 1/hx^2 = 0.25)
    jacobi2(e1, e2, e1b, e2b, J2, g2, 512, 512, 0.25f, 0);
    launch_cdt(e1, e2, J2, g2, r2u, r2v, 512, 512, 0.25f, 2);
    { dim3 gr((256 + 255) / 256, 256, Bt);
      restrict_kernel<<<gr, 256, 0, stream>>>(J2, J2 + 2 * HW2, J2 + 6 * HW2,
                                              r2u, r2v, g2, J3, g3, 256, 256); }
    zero(e31, Bt * HW3); zero(e32, Bt * HW3);
    // depth 3 == max (256^2, hx=4): parabolic smoothing
    jacobi2(e31, e32, e31b, e32b, J3, g3, 256, 256, 0.0625f, 1);
    // prolong L3 -> L2, post-smooth
    { dim3 gu((512 + 255) / 256, 512, Bt);
      upsample_add<<<gu, 256, 0, stream>>>(e31, e1, 512, 512);
      upsample_add<<<gu, 256, 0, stream>>>(e32, e2, 512, 512); }
    jacobi2(e1, e2, e1b, e2b, J2, g2, 512, 512, 0.25f, 0);
    // prolong L2 -> L1, post-smooth
    { dim3 gu((W + 255) / 256, H, Bt);
      upsample_add<<<gu, 256, 0, stream>>>(e1, u, H, W);
      upsample_add<<<gu, 256, 0, stream>>>(e2, v, H, W); }
    jacobi2(u, v, u2, v2, J1, g1, H, W, 1.0f, 0);
  }

  hipMemcpyAsync(out, u, Bt * HW * sizeof(float), hipMemcpyDeviceToDevice, stream);
  hipMemcpyAsync(out + Bt * HW, v, Bt * HW * sizeof(float), hipMemcpyDeviceToDevice, stream);
}